// MAMLModel_8100308320891
// MI455X (gfx1250) — compile-verified
//
#include <hip/hip_runtime.h>
#include <hip/hip_bf16.h>

typedef __attribute__((ext_vector_type(2))) float v2f;
typedef __attribute__((ext_vector_type(8))) float v8f;

#define HH 2
#define CC 16
#define HC 32
#define GG 64

// ---------- monotone float<->uint key for atomic max over signed floats ----------
__device__ __forceinline__ unsigned ordf(float f) {
  unsigned u = __float_as_uint(f);
  return (u & 0x80000000u) ? ~u : (u | 0x80000000u);
}
__device__ __forceinline__ float unordf(unsigned u) {
  return (u & 0x80000000u) ? __uint_as_float(u & 0x7FFFFFFFu) : __uint_as_float(~u);
}

// ---------- fills ----------
__global__ void k_fill_f(float* __restrict__ p, float v, size_t n) {
  size_t i = (size_t)blockIdx.x * blockDim.x + threadIdx.x;
  if (i < n) p[i] = v;
}
__global__ void k_fill_u(unsigned* __restrict__ p, unsigned v, size_t n) {
  size_t i = (size_t)blockIdx.x * blockDim.x + threadIdx.x;
  if (i < n) p[i] = v;
}

// ---------- edge_attr mean (sum reduction) ----------
__global__ void k_reduce_sum(const float* __restrict__ x, int n, float* __restrict__ out) {
  __shared__ float sd[256];
  float s = 0.f;
  for (int i = blockIdx.x * blockDim.x + threadIdx.x; i < n; i += gridDim.x * blockDim.x)
    s += x[i];
  sd[threadIdx.x] = s;
  __syncthreads();
  for (int o = 128; o > 0; o >>= 1) {
    if ((int)threadIdx.x < o) sd[threadIdx.x] += sd[threadIdx.x + o];
    __syncthreads();
  }
  if (threadIdx.x == 0) atomicAdd(out, sd[0]);
}

// scal[0]=ea_sum -> scal[5]=ea_mean; scal[1..2]=wecoef layer1; scal[3..4]=wecoef layer2
__global__ void k_prep(float* __restrict__ scal,
                       const float* __restrict__ We1, const float* __restrict__ ae1,
                       const float* __restrict__ We2, const float* __restrict__ ae2, int E) {
  if (threadIdx.x == 0 && blockIdx.x == 0) {
    scal[5] = scal[0] / (float)E;
    for (int h = 0; h < HH; ++h) {
      float s1 = 0.f, s2 = 0.f;
      for (int c = 0; c < CC; ++c) {
        s1 += We1[h * CC + c] * ae1[h * CC + c];
        s2 += We2[h * CC + c] * ae2[h * CC + c];
      }
      scal[1 + h] = s1;
      scal[3 + h] = s2;
    }
  }
}

// ---------- fp32 WMMA GEMM: out[M,32] = A[M,K] @ W[K,32]  (row major) ----------
// one wave -> one 16x32 output tile via V_WMMA_F32_16X16X4_F32, K-stepped by 4
__global__ void k_gemm_wmma(const float* __restrict__ A, const float* __restrict__ W,
                            float* __restrict__ out, int M, int K) {
  int wave = threadIdx.x >> 5;
  int lane = threadIdx.x & 31;
  int tile = blockIdx.x * (blockDim.x >> 5) + wave;
  if (tile * 16 >= M) return;  // wave-uniform: EXEC stays all-ones for WMMA
  int row0 = tile * 16;
  int ml = lane & 15;           // A row / B col within tile
  int kb = (lane >> 4) * 2;     // K sub-offset held by this half-wave
  v8f acc0 = {}; v8f acc1 = {};
  for (int k0 = 0; k0 < K; k0 += 4) {
    const float* ap = A + (size_t)(row0 + ml) * K + (k0 + kb);
    v2f a; a.x = ap[0]; a.y = ap[1];
    const float* wp = W + (size_t)(k0 + kb) * HC + ml;
    v2f b0; b0.x = wp[0];  b0.y = wp[HC];        // cols 0..15
    v2f b1; b1.x = wp[16]; b1.y = wp[HC + 16];   // cols 16..31
    acc0 = __builtin_amdgcn_wmma_f32_16x16x4_f32(false, a, false, b0, (short)0, acc0, false, false);
    acc1 = __builtin_amdgcn_wmma_f32_16x16x4_f32(false, a, false, b1, (short)0, acc1, false, false);
  }
  int orow = row0 + ((lane >> 4) * 8);
  float* op = out + (size_t)orow * HC + ml;
#pragma unroll
  for (int v = 0; v < 8; ++v) {
    op[(size_t)v * HC] = acc0[v];
    op[(size_t)v * HC + 16] = acc1[v];
  }
}

// ---------- per-node attention scalars: asrc/adst[n,h] = sum_c h*a ----------
__global__ void k_node_att(const float* __restrict__ h, const float* __restrict__ a_src,
                           const float* __restrict__ a_dst,
                           float* __restrict__ asrc, float* __restrict__ adst, int N) {
  int n = blockIdx.x * blockDim.x + threadIdx.x;
  if (n >= N) return;
  const float4* h4 = (const float4*)(h + (size_t)n * HC);   // rows are 128B aligned
  const float4* s4 = (const float4*)a_src;
  const float4* d4 = (const float4*)a_dst;
  for (int hh = 0; hh < HH; ++hh) {
    float s = 0.f, d = 0.f;
#pragma unroll
    for (int q = 0; q < 4; ++q) {
      float4 hv = h4[hh * 4 + q];
      float4 sv = s4[hh * 4 + q];
      float4 dv = d4[hh * 4 + q];
      s += hv.x * sv.x + hv.y * sv.y + hv.z * sv.z + hv.w * sv.w;
      d += hv.x * dv.x + hv.y * dv.y + hv.z * dv.z + hv.w * dv.w;
    }
    asrc[n * HH + hh] = s;
    adst[n * HH + hh] = d;
  }
}

// ---------- edge pass 1 (one thread per edge*head): alpha + segment max ----------
__global__ void k_edge_max(const int* __restrict__ src, const int* __restrict__ dst,
                           const float* __restrict__ ea, const float* __restrict__ scal,
                           int wecoff, const float* __restrict__ asrc,
                           const float* __restrict__ adst, float* __restrict__ alpha,
                           unsigned* __restrict__ mkey, int E, int Etot) {
  int idx = blockIdx.x * blockDim.x + threadIdx.x;
  if (idx >= Etot * HH) return;
  int e = idx >> 1, hh = idx & 1;
  int s, d;
  float eav;
  if (e < E) { s = src[e]; d = dst[e]; eav = ea[e]; }
  else       { s = d = e - E; eav = scal[5]; }       // self-loop, mean edge attr
  float al = asrc[s * HH + hh] + adst[d * HH + hh] + eav * scal[wecoff + hh];
  al = al > 0.f ? al : 0.2f * al;                     // leaky_relu(0.2)
  alpha[(size_t)e * HH + hh] = al;
  atomicMax(&mkey[d * HH + hh], ordf(al));
}

// ---------- edge pass 2 (one thread per edge*head): ex, sums, messages ----------
__global__ void k_edge_accum(const int* __restrict__ src, const int* __restrict__ dst,
                             const float* __restrict__ alpha, const unsigned* __restrict__ mkey,
                             const float* __restrict__ hfeat, float* __restrict__ ssum,
                             float* __restrict__ oacc, int E, int Etot) {
  int idx = blockIdx.x * blockDim.x + threadIdx.x;
  if (idx >= Etot * HH) return;
  int e = idx >> 1, hh = idx & 1;
  int s, d;
  if (e < E) { s = src[e]; d = dst[e]; } else { s = d = e - E; }
  float m = unordf(mkey[d * HH + hh]);
  float ex = expf(alpha[(size_t)e * HH + hh] - m);
  atomicAdd(&ssum[d * HH + hh], ex);
  // gather one head (64B) of h[src] as 4x b128, scatter 16 float atomics
  const float4* hs4 = (const float4*)(hfeat + (size_t)s * HC + hh * CC);
  float4 h0 = hs4[0], h1 = hs4[1], h2 = hs4[2], h3 = hs4[3];
  float* od = oacc + (size_t)d * HC + hh * CC;
  atomicAdd(&od[0],  h0.x * ex); atomicAdd(&od[1],  h0.y * ex);
  atomicAdd(&od[2],  h0.z * ex); atomicAdd(&od[3],  h0.w * ex);
  atomicAdd(&od[4],  h1.x * ex); atomicAdd(&od[5],  h1.y * ex);
  atomicAdd(&od[6],  h1.z * ex); atomicAdd(&od[7],  h1.w * ex);
  atomicAdd(&od[8],  h2.x * ex); atomicAdd(&od[9],  h2.y * ex);
  atomicAdd(&od[10], h2.z * ex); atomicAdd(&od[11], h2.w * ex);
  atomicAdd(&od[12], h3.x * ex); atomicAdd(&od[13], h3.y * ex);
  atomicAdd(&od[14], h3.z * ex); atomicAdd(&od[15], h3.w * ex);
}

// ---------- normalize + bias (+optional relu), in place, vectorized ----------
__global__ void k_node_final(float* __restrict__ oacc, const float* __restrict__ ssum,
                             const float* __restrict__ bias, int relu, int N) {
  int n = blockIdx.x * blockDim.x + threadIdx.x;
  if (n >= N) return;
  float4* o4 = (float4*)(oacc + (size_t)n * HC);
  const float4* b4 = (const float4*)bias;
  for (int hh = 0; hh < HH; ++hh) {
    float inv = 1.0f / (ssum[n * HH + hh] + 1e-16f);
#pragma unroll
    for (int q = 0; q < 4; ++q) {
      float4 v = o4[hh * 4 + q];
      float4 b = b4[hh * 4 + q];
      v.x = v.x * inv + b.x; v.y = v.y * inv + b.y;
      v.z = v.z * inv + b.z; v.w = v.w * inv + b.w;
      if (relu) {
        v.x = fmaxf(v.x, 0.f); v.y = fmaxf(v.y, 0.f);
        v.z = fmaxf(v.z, 0.f); v.w = fmaxf(v.w, 0.f);
      }
      o4[hh * 4 + q] = v;
    }
  }
}

// ---------- global mean pool: LDS-staged per-block, then global atomics ----------
__global__ void k_pool(const float* __restrict__ feat, const int* __restrict__ batch,
                       float* __restrict__ pool, float* __restrict__ cnt, int N) {
  __shared__ float sp[GG * HC];
  __shared__ float sc[GG];
  for (int i = threadIdx.x; i < GG * HC; i += blockDim.x) sp[i] = 0.f;
  for (int i = threadIdx.x; i < GG; i += blockDim.x) sc[i] = 0.f;
  __syncthreads();
  int n = blockIdx.x * blockDim.x + threadIdx.x;
  if (n < N) {
    int g = batch[n];
    const float4* f4 = (const float4*)(feat + (size_t)n * HC);
#pragma unroll
    for (int q = 0; q < 8; ++q) {
      float4 v = f4[q];
      atomicAdd(&sp[g * HC + q * 4 + 0], v.x);
      atomicAdd(&sp[g * HC + q * 4 + 1], v.y);
      atomicAdd(&sp[g * HC + q * 4 + 2], v.z);
      atomicAdd(&sp[g * HC + q * 4 + 3], v.w);
    }
    atomicAdd(&sc[g], 1.f);
  }
  __syncthreads();
  for (int i = threadIdx.x; i < GG * HC; i += blockDim.x)
    if (sc[i / HC] > 0.f) atomicAdd(&pool[i], sp[i]);
  for (int i = threadIdx.x; i < GG; i += blockDim.x)
    if (sc[i] > 0.f) atomicAdd(&cnt[i], sc[i]);
}

// ---------- predictor MLP: 32 -> relu(32) -> 2, one thread per graph ----------
__global__ void k_mlp(const float* __restrict__ pool, const float* __restrict__ cnt,
                      const float* __restrict__ Wf1, const float* __restrict__ bf1,
                      const float* __restrict__ Wf2, const float* __restrict__ bf2,
                      float* __restrict__ out) {
  int g = threadIdx.x;
  if (g >= GG) return;
  float c = fmaxf(cnt[g], 1.f);
  float emb[HC];
  for (int j = 0; j < HC; ++j) emb[j] = pool[g * HC + j] / c;
  float t[32];
  for (int k = 0; k < 32; ++k) {
    float s = bf1[k];
    for (int j = 0; j < HC; ++j) s += emb[j] * Wf1[j * 32 + k];
    t[k] = fmaxf(s, 0.f);
  }
  for (int o = 0; o < 2; ++o) {
    float s = bf2[o];
    for (int k = 0; k < 32; ++k) s += t[k] * Wf2[k * 2 + o];
    out[g * 2 + o] = s;
  }
}

extern "C" void kernel_launch(void* const* d_in, const int* in_sizes, int n_in,
                              void* d_out, int out_size, void* d_ws, size_t ws_size,
                              hipStream_t stream) {
  const float* x    = (const float*)d_in[0];
  const int*   ei   = (const int*)d_in[1];
  const float* ea   = (const float*)d_in[2];
  const int*   bat  = (const int*)d_in[3];
  const float* W1   = (const float*)d_in[4];
  const float* as1  = (const float*)d_in[5];
  const float* ad1  = (const float*)d_in[6];
  const float* We1  = (const float*)d_in[7];
  const float* ae1  = (const float*)d_in[8];
  const float* b1   = (const float*)d_in[9];
  const float* W2   = (const float*)d_in[10];
  const float* as2  = (const float*)d_in[11];
  const float* ad2  = (const float*)d_in[12];
  const float* We2  = (const float*)d_in[13];
  const float* ae2  = (const float*)d_in[14];
  const float* b2   = (const float*)d_in[15];
  const float* Wf1  = (const float*)d_in[16];
  const float* bf1  = (const float*)d_in[17];
  const float* Wf2  = (const float*)d_in[18];
  const float* bf2  = (const float*)d_in[19];
  float* out = (float*)d_out;

  const int N = in_sizes[0] / 128;     // 100000
  const int E = in_sizes[1] / 2;       // 3200000
  const int Etot = E + N;
  const int* src = ei;
  const int* dstp = ei + E;

  // workspace carve-out (floats); all row bases 64B-aligned by construction
  float* ws = (float*)d_ws;
  float*    scal  = ws;                                 // 64 slots
  float*    hbuf  = ws + 64;                            // N*32
  float*    obuf  = hbuf + (size_t)N * HC;              // N*32
  float*    asrc  = obuf + (size_t)N * HC;              // N*2
  float*    adst  = asrc + (size_t)N * HH;              // N*2
  unsigned* mkey  = (unsigned*)(adst + (size_t)N * HH); // N*2
  float*    ssum  = (float*)(mkey + (size_t)N * HH);    // N*2
  float*    alpha = ssum + (size_t)N * HH;              // Etot*2
  float*    pools = alpha + (size_t)Etot * HH;          // 64*32
  float*    cnts  = pools + GG * HC;                    // 64

  const unsigned ORD_NEG_INF = 0x007FFFFFu; // ordf(-inf)
  dim3 B(256);
  auto gr = [](size_t n) { return dim3((unsigned)((n + 255) / 256)); };
  int gtiles = (N + 15) / 16;
  dim3 gemm_grid((gtiles + 7) / 8);

  // edge_attr mean + per-head edge coefficients
  k_fill_f<<<1, 64, 0, stream>>>(scal, 0.f, 64);
  k_reduce_sum<<<512, 256, 0, stream>>>(ea, E, scal);
  k_prep<<<1, 32, 0, stream>>>(scal, We1, ae1, We2, ae2, E);

  // ---- layer 1 ----
  k_gemm_wmma<<<gemm_grid, B, 0, stream>>>(x, W1, hbuf, N, 128);
  k_node_att<<<gr(N), B, 0, stream>>>(hbuf, as1, ad1, asrc, adst, N);
  k_fill_u<<<gr((size_t)N * HH), B, 0, stream>>>(mkey, ORD_NEG_INF, (size_t)N * HH);
  k_fill_f<<<gr((size_t)N * HH), B, 0, stream>>>(ssum, 0.f, (size_t)N * HH);
  k_fill_f<<<gr((size_t)N * HC), B, 0, stream>>>(obuf, 0.f, (size_t)N * HC);
  k_edge_max<<<gr((size_t)Etot * HH), B, 0, stream>>>(src, dstp, ea, scal, 1, asrc, adst, alpha, mkey, E, Etot);
  k_edge_accum<<<gr((size_t)Etot * HH), B, 0, stream>>>(src, dstp, alpha, mkey, hbuf, ssum, obuf, E, Etot);
  k_node_final<<<gr(N), B, 0, stream>>>(obuf, ssum, b1, /*relu=*/1, N);

  // ---- layer 2 ----
  k_gemm_wmma<<<gemm_grid, B, 0, stream>>>(obuf, W2, hbuf, N, 32);
  k_node_att<<<gr(N), B, 0, stream>>>(hbuf, as2, ad2, asrc, adst, N);
  k_fill_u<<<gr((size_t)N * HH), B, 0, stream>>>(mkey, ORD_NEG_INF, (size_t)N * HH);
  k_fill_f<<<gr((size_t)N * HH), B, 0, stream>>>(ssum, 0.f, (size_t)N * HH);
  k_fill_f<<<gr((size_t)N * HC), B, 0, stream>>>(obuf, 0.f, (size_t)N * HC);
  k_edge_max<<<gr((size_t)Etot * HH), B, 0, stream>>>(src, dstp, ea, scal, 3, asrc, adst, alpha, mkey, E, Etot);
  k_edge_accum<<<gr((size_t)Etot * HH), B, 0, stream>>>(src, dstp, alpha, mkey, hbuf, ssum, obuf, E, Etot);
  k_node_final<<<gr(N), B, 0, stream>>>(obuf, ssum, b2, /*relu=*/0, N);

  // ---- pool + MLP ----
  k_fill_f<<<gr(GG * HC + GG), B, 0, stream>>>(pools, 0.f, GG * HC + GG); // pools & cnts contiguous
  k_pool<<<gr(N), B, 0, stream>>>(obuf, bat, pools, cnts, N);
  k_mlp<<<1, 64, 0, stream>>>(pools, cnts, Wf1, bf1, Wf2, bf2, out);
}